// MolecularGVAE_50921132261983
// MI455X (gfx1250) — compile-verified
//
#include <hip/hip_runtime.h>
#include <hip/hip_bf16.h>

// ---------------------------------------------------------------------------
// MolecularGVAE for gfx1250 (MI455X). fp32 throughout (matches reference).
// Dense node-level GEMMs use V_WMMA_F32_16X16X4_F32 with LDS-staged panels
// (branch-free inner loop: ds_load_b64 x2 + wmma). The O(N^2) edge predictor
// is strength-reduced: ei@We1 = z_i@We1_top + z_j@We1_bot, and LayerNorm->dot
// folds into 3 running sums per pair.
// ---------------------------------------------------------------------------

#define NN   1536
#define EE   49152
#define FIN  27
#define HH   64
#define LL   32

typedef float v2f __attribute__((ext_vector_type(2)));
typedef float v8f __attribute__((ext_vector_type(8)));

// ---------------- WMMA fp32 GEMM: C[M,Nc] = A[M,K] @ B[K,Nc] (+bias, ?relu)
// Block = 4 waves -> 64x16 C tile. A[64,Kp] and B^T[16,Kp] staged in LDS with
// K zero-padded to Kp (mult of 4), so the wmma loop has no bounds checks.
// Requires M % 64 == 0 is NOT assumed (staging+store guarded), K <= 92.
#define AP 94   // LDS pitch (floats): >= 92, even (keeps b64 alignment)
__global__ __launch_bounds__(128)
void gemm_wmma_f32(const float* __restrict__ A, const float* __restrict__ B,
                   const float* __restrict__ bias, float* __restrict__ C,
                   int M, int Nc, int K, int relu) {
    __shared__ float sA[64 * AP];   // A panel, row-major, pitch AP
    __shared__ float sBt[16 * AP];  // B panel transposed: [col][k]

    const int t    = threadIdx.x;          // 0..127
    const int Kp   = (K + 3) & ~3;
    const int row0 = blockIdx.x * 64;
    const int col0 = blockIdx.y * 16;

    // ---- stage A: thread pair per row, stride-2 over k
    {
        const int r  = t >> 1;             // 0..63
        const int gr = row0 + r;
        const bool rok = (gr < M);
        for (int k = (t & 1); k < Kp; k += 2)
            sA[r * AP + k] = (rok && k < K) ? A[gr * K + k] : 0.0f;
    }
    // ---- stage B^T: 16 cols x Kp, 8 threads stride over k per col
    {
        const int c  = t & 15;
        const bool cok = (col0 + c) < Nc;
        for (int k = (t >> 4); k < Kp; k += 8)
            sBt[c * AP + k] = (cok && k < K) ? B[k * Nc + col0 + c] : 0.0f;
    }
    __syncthreads();

    const int lane = t & 31;
    const int wv   = t >> 5;               // wave id 0..3 -> 16-row slice
    const int half = lane >> 4;            // K sub-pair {0,1} vs {2,3}
    const int l    = lane & 15;

    const float* pA = &sA[(wv * 16 + l) * AP + (half << 1)];
    const float* pB = &sBt[l * AP + (half << 1)];

    v8f c = {};
    #pragma unroll 2
    for (int k0 = 0; k0 < Kp; k0 += 4) {
        const v2f a = *(const v2f*)(pA + k0);   // ds_load_b64
        const v2f b = *(const v2f*)(pB + k0);   // ds_load_b64
        c = __builtin_amdgcn_wmma_f32_16x16x4_f32(false, a, false, b,
                                                  (short)0, c, false, false);
    }

    // C/D layout: lane l<16 -> N=l, M=v ; lane>=16 -> N=l, M=8+v
    const int bc = col0 + l;
    if (bc < Nc) {
        const float bv = bias ? bias[bc] : 0.0f;
        #pragma unroll
        for (int v = 0; v < 8; ++v) {
            const int m = row0 + wv * 16 + v + half * 8;
            if (m < M) {
                float val = c[v] + bv;
                if (relu) val = fmaxf(val, 0.0f);
                C[m * Nc + bc] = val;
            }
        }
    }
}

// ---------------- GATv2 pass A: per-edge logit + segment-max via u32 atomicMax
__device__ __forceinline__ unsigned f2key(float f) {
    unsigned b = __float_as_uint(f);
    return (b & 0x80000000u) ? ~b : (b | 0x80000000u);
}
__device__ __forceinline__ float key2f(unsigned k) {
    return __uint_as_float((k & 0x80000000u) ? (k & 0x7FFFFFFFu) : ~k);
}

__global__ void gat_edge_max(const float* __restrict__ xl, const float* __restrict__ xr,
                             const float* __restrict__ att,
                             const int* __restrict__ srcE, const int* __restrict__ dstE,
                             int E, int n, float* __restrict__ sbuf,
                             unsigned* __restrict__ mkey) {
    const int e = blockIdx.x * blockDim.x + threadIdx.x;
    const int Etot = E + n;
    if (e >= Etot) return;
    const int s = (e < E) ? srcE[e] : (e - E);
    const int d = (e < E) ? dstE[e] : (e - E);
    const float4* pl = (const float4*)(xl + s * HH);
    const float4* pr = (const float4*)(xr + d * HH);
    const float4* pa = (const float4*)att;
    float acc = 0.0f;
    #pragma unroll 4
    for (int h = 0; h < HH / 4; ++h) {
        const float4 a = pl[h], r = pr[h], w = pa[h];
        float v;
        v = a.x + r.x; v = (v > 0.0f) ? v : 0.2f * v; acc = fmaf(w.x, v, acc);
        v = a.y + r.y; v = (v > 0.0f) ? v : 0.2f * v; acc = fmaf(w.y, v, acc);
        v = a.z + r.z; v = (v > 0.0f) ? v : 0.2f * v; acc = fmaf(w.z, v, acc);
        v = a.w + r.w; v = (v > 0.0f) ? v : 0.2f * v; acc = fmaf(w.w, v, acc);
    }
    sbuf[e] = acc;
    atomicMax(mkey + d, f2key(acc));
}

// ---------------- GATv2 pass B: w=exp(s-m); denom += w; acc[dst] += w*xl[src]
__global__ __launch_bounds__(HH)
void gat_edge_accum(const float* __restrict__ xl, const float* __restrict__ sbuf,
                    const unsigned* __restrict__ mkey,
                    const int* __restrict__ srcE, const int* __restrict__ dstE,
                    int E, int n, float* __restrict__ denom, float* __restrict__ acc) {
    const int e = blockIdx.x;
    const int h = threadIdx.x;
    const int s = (e < E) ? srcE[e] : (e - E);
    const int d = (e < E) ? dstE[e] : (e - E);
    const float w = expf(sbuf[e] - key2f(mkey[d]));
    if (h == 0) atomicAdd(denom + d, w);
    atomicAdd(acc + d * HH + h, w * xl[s * HH + h]);
}

// ---------------- GATv2 finalize: out = relu(acc/denom + bias)
__global__ __launch_bounds__(HH)
void gat_finalize(const float* __restrict__ acc, const float* __restrict__ denom,
                  const float* __restrict__ bias, float* __restrict__ out) {
    const int i = blockIdx.x, h = threadIdx.x;
    const float v = acc[i * HH + h] / denom[i] + bias[h];
    out[i * HH + h] = fmaxf(v, 0.0f);
}

// ---------------- LayerNorm over rows of length 64 (in place)
__global__ __launch_bounds__(HH)
void layernorm64(float* __restrict__ buf, const float* __restrict__ g,
                 const float* __restrict__ b) {
    __shared__ float s1[HH], s2[HH];
    const int i = blockIdx.x, t = threadIdx.x;
    const float x = buf[i * HH + t];
    s1[t] = x; s2[t] = x * x;
    __syncthreads();
    for (int off = 32; off > 0; off >>= 1) {
        if (t < off) { s1[t] += s1[t + off]; s2[t] += s2[t + off]; }
        __syncthreads();
    }
    const float mean = s1[0] * (1.0f / HH);
    const float var  = s2[0] * (1.0f / HH) - mean * mean;
    const float inv  = rsqrtf(var + 1e-5f);
    buf[i * HH + t] = (x - mean) * inv * g[t] + b[t];
}

// ---------------- z = mu + eps * exp(0.5*logvar)
__global__ void reparam(const float* __restrict__ mu, const float* __restrict__ lv,
                        const float* __restrict__ eps, float* __restrict__ z, int n) {
    const int i = blockIdx.x * blockDim.x + threadIdx.x;
    if (i < n) z[i] = fmaf(eps[i], expf(0.5f * lv[i]), mu[i]);
}

// ---------------- graph head: gh = mean(h2, axis=0); gmu/glv = gh@W + b
__global__ __launch_bounds__(HH)
void graph_head(const float* __restrict__ h2,
                const float* __restrict__ Wgm, const float* __restrict__ bgm,
                const float* __restrict__ Wgv, const float* __restrict__ bgv,
                float* __restrict__ gmu, float* __restrict__ glv, int n) {
    __shared__ float gh[HH];
    const int t = threadIdx.x;
    float s = 0.0f;
    for (int i = 0; i < n; ++i) s += h2[i * HH + t];
    gh[t] = s / (float)n;
    __syncthreads();
    if (t < LL) {
        float a = 0.0f, b = 0.0f;
        #pragma unroll 8
        for (int k = 0; k < HH; ++k) {
            a = fmaf(gh[k], Wgm[k * LL + t], a);
            b = fmaf(gh[k], Wgv[k * LL + t], b);
        }
        gmu[t] = a + bgm[t];
        glv[t] = b + bgv[t];
    }
}

// ---------------- cat = [nf3 (27) | h2 (64)]  -> [N,91]
__global__ void concat_kernel(const float* __restrict__ nf, const float* __restrict__ h,
                              float* __restrict__ cat, int n) {
    const int idx = blockIdx.x * blockDim.x + threadIdx.x;
    const int total = n * (FIN + HH);
    if (idx >= total) return;
    const int i = idx / (FIN + HH), k = idx % (FIN + HH);
    cat[idx] = (k < FIN) ? nf[i * FIN + k] : h[i * HH + (k - FIN)];
}

// ---------------- fold LN gain/bias and We2 into per-channel constants
__global__ __launch_bounds__(HH)
void edge_consts(const float* __restrict__ g, const float* __restrict__ b,
                 const float* __restrict__ We2, const float* __restrict__ be2,
                 float* __restrict__ c) {
    __shared__ float sg[HH], sb[HH];
    const int t = threadIdx.x;
    const float w  = We2[t];
    const float gw = g[t] * w;
    c[t] = gw;
    sg[t] = gw; sb[t] = b[t] * w;
    __syncthreads();
    for (int off = 32; off > 0; off >>= 1) {
        if (t < off) { sg[t] += sg[t + off]; sb[t] += sb[t + off]; }
        __syncthreads();
    }
    if (t == 0) { c[HH] = sg[0]; c[HH + 1] = sb[0] + be2[0]; }
}

// ---------------- all-pairs edge predictor, strength reduced
// logit(i,j) = inv_std * (sum_k r_k*gw_k - mean*sum_gw) + c0,
//   r = relu(ZA[i] + ZB[j]),  be1 folded into ZA.
#define TI 16
#define TJ 16
#define LDP 68   // LDS row pitch (floats): mult of 4 (b128 align) + bank stagger
__global__ __launch_bounds__(256)
void edge_pairs(const float* __restrict__ ZA, const float* __restrict__ ZB,
                const float* __restrict__ consts, float* __restrict__ logits, int n) {
    const int i0 = blockIdx.x * TI;
    const int j0 = blockIdx.y * TJ;
    if (j0 + TJ - 1 <= i0) return;          // tile entirely in lower triangle
    __shared__ float sA[TI * LDP];
    __shared__ float sB[TJ * LDP];
    __shared__ float sgw[HH];
    const int t = threadIdx.x;              // 256
    for (int idx = t; idx < TI * HH; idx += 256) {
        const int r = idx >> 6, k = idx & 63;
        const int gi = i0 + r, gj = j0 + r;
        sA[r * LDP + k] = (gi < n) ? ZA[gi * HH + k] : 0.0f;
        sB[r * LDP + k] = (gj < n) ? ZB[gj * HH + k] : 0.0f;
    }
    if (t < HH) sgw[t] = consts[t];
    __syncthreads();
    const float sum_gw = consts[HH];
    const float c0     = consts[HH + 1];
    const int ti = t >> 4, tj = t & 15;
    const int i = i0 + ti, j = j0 + tj;
    if (i < n && j < n && j > i) {
        const float4* pa = (const float4*)&sA[ti * LDP];
        const float4* pb = (const float4*)&sB[tj * LDP];
        const float4* pg = (const float4*)sgw;
        float sr = 0.0f, sr2 = 0.0f, sg = 0.0f;
        #pragma unroll 4
        for (int k = 0; k < HH / 4; ++k) {
            const float4 va = pa[k], vb = pb[k], vg = pg[k];
            float v;
            v = fmaxf(va.x + vb.x, 0.0f); sr += v; sr2 = fmaf(v, v, sr2); sg = fmaf(v, vg.x, sg);
            v = fmaxf(va.y + vb.y, 0.0f); sr += v; sr2 = fmaf(v, v, sr2); sg = fmaf(v, vg.y, sg);
            v = fmaxf(va.z + vb.z, 0.0f); sr += v; sr2 = fmaf(v, v, sr2); sg = fmaf(v, vg.z, sg);
            v = fmaxf(va.w + vb.w, 0.0f); sr += v; sr2 = fmaf(v, v, sr2); sg = fmaf(v, vg.w, sg);
        }
        const float mean = sr * (1.0f / HH);
        const float var  = sr2 * (1.0f / HH) - mean * mean;
        const float inv  = rsqrtf(var + 1e-5f);
        const float logit = inv * (sg - mean * sum_gw) + c0;
        const int p = i * n - (i * (i + 1)) / 2 + (j - i - 1);
        logits[p] = logit;
    }
}

// ---------------------------------------------------------------------------
extern "C" void kernel_launch(void* const* d_in, const int* in_sizes, int n_in,
                              void* d_out, int out_size, void* d_ws, size_t ws_size,
                              hipStream_t stream) {
    (void)in_sizes; (void)n_in; (void)out_size; (void)ws_size;
    const int n = NN, E = EE, Etot = EE + NN;

    // ---- inputs (setup_inputs dict order)
    const float* x     = (const float*)d_in[0];
    const int*   eidx  = (const int*)  d_in[1];
    const float* epsN  = (const float*)d_in[2];
    // d_in[3] = eps_graph (z_graph unused downstream)
    const float* Wl1 = (const float*)d_in[4],  *bl1 = (const float*)d_in[5];
    const float* Wr1 = (const float*)d_in[6],  *br1 = (const float*)d_in[7];
    const float* att1= (const float*)d_in[8],  *bias1=(const float*)d_in[9];
    const float* Wl2 = (const float*)d_in[10], *bl2 = (const float*)d_in[11];
    const float* Wr2 = (const float*)d_in[12], *br2 = (const float*)d_in[13];
    const float* att2= (const float*)d_in[14], *bias2=(const float*)d_in[15];
    const float* Wnm = (const float*)d_in[16], *bnm = (const float*)d_in[17];
    const float* Wnv = (const float*)d_in[18], *bnv = (const float*)d_in[19];
    const float* Wgm = (const float*)d_in[20], *bgm = (const float*)d_in[21];
    const float* Wgv = (const float*)d_in[22], *bgv = (const float*)d_in[23];
    const float* Wd1 = (const float*)d_in[24], *bd1 = (const float*)d_in[25];
    const float* ln1g= (const float*)d_in[26], *ln1b= (const float*)d_in[27];
    const float* Wd2 = (const float*)d_in[28], *bd2 = (const float*)d_in[29];
    const float* ln2g= (const float*)d_in[30], *ln2b= (const float*)d_in[31];
    const float* Wd3 = (const float*)d_in[32], *bd3 = (const float*)d_in[33];
    const float* We1 = (const float*)d_in[34], *be1 = (const float*)d_in[35];
    const float* lneg= (const float*)d_in[36], *lneb= (const float*)d_in[37];
    const float* We2 = (const float*)d_in[38], *be2 = (const float*)d_in[39];
    const float* Wf1 = (const float*)d_in[40], *bf1 = (const float*)d_in[41];
    const float* lnfg= (const float*)d_in[42], *lnfb= (const float*)d_in[43];
    const float* Wf2 = (const float*)d_in[44], *bf2 = (const float*)d_in[45];

    const int* srcE = eidx;
    const int* dstE = eidx + E;

    // ---- output layout (return-order concat)
    const long long P = (long long)n * (n - 1) / 2;
    float* out      = (float*)d_out;
    float* o_nf     = out;                          // [N,27]
    float* o_el     = out + (long long)n * FIN;     // [P]
    float* o_nmu    = o_el + P;                     // [N,32]
    float* o_nlv    = o_nmu + (long long)n * LL;    // [N,32]
    float* o_gmu    = o_nlv + (long long)n * LL;    // [32]
    float* o_glv    = o_gmu + LL;                   // [32]

    // ---- workspace carving (floats, 64-float aligned)
    float* ws = (float*)d_ws;
    size_t off = 0;
    auto carve = [&](size_t cnt) { float* p = ws + off; off += (cnt + 63) & ~(size_t)63; return p; };
    float*    xl    = carve((size_t)n * HH);
    float*    xr    = carve((size_t)n * HH);
    float*    sbuf  = carve((size_t)Etot);
    unsigned* mkey  = (unsigned*)carve((size_t)n);
    float*    denom = carve((size_t)n);
    float*    acc   = carve((size_t)n * HH);
    float*    h1    = carve((size_t)n * HH);
    float*    h2    = carve((size_t)n * HH);
    float*    z     = carve((size_t)n * LL);
    float*    bufA  = carve((size_t)n * HH);
    float*    bufB  = carve((size_t)n * HH);
    float*    nf3   = carve((size_t)n * FIN);
    float*    cat   = carve((size_t)n * (FIN + HH));
    float*    csts  = carve((size_t)(HH + 2));

    auto gemm = [&](const float* A, const float* B, const float* bias, float* C,
                    int M, int Nc, int K, int relu) {
        dim3 g((M + 63) / 64, (Nc + 15) / 16);
        gemm_wmma_f32<<<g, 128, 0, stream>>>(A, B, bias, C, M, Nc, K, relu);
    };
    auto zero_seg = [&]() {
        hipMemsetAsync(mkey,  0, (size_t)n * sizeof(unsigned), stream);
        hipMemsetAsync(denom, 0, (size_t)n * sizeof(float),    stream);
        hipMemsetAsync(acc,   0, (size_t)n * HH * sizeof(float), stream);
    };
    const int EB = (Etot + 255) / 256;

    // ================= GATv2 layer 1 =================
    zero_seg();
    gemm(x, Wl1, bl1, xl, n, HH, FIN, 0);
    gemm(x, Wr1, br1, xr, n, HH, FIN, 0);
    gat_edge_max  <<<EB, 256, 0, stream>>>(xl, xr, att1, srcE, dstE, E, n, sbuf, mkey);
    gat_edge_accum<<<Etot, HH, 0, stream>>>(xl, sbuf, mkey, srcE, dstE, E, n, denom, acc);
    gat_finalize  <<<n, HH, 0, stream>>>(acc, denom, bias1, h1);

    // ================= GATv2 layer 2 =================
    zero_seg();
    gemm(h1, Wl2, bl2, xl, n, HH, HH, 0);
    gemm(h1, Wr2, br2, xr, n, HH, HH, 0);
    gat_edge_max  <<<EB, 256, 0, stream>>>(xl, xr, att2, srcE, dstE, E, n, sbuf, mkey);
    gat_edge_accum<<<Etot, HH, 0, stream>>>(xl, sbuf, mkey, srcE, dstE, E, n, denom, acc);
    gat_finalize  <<<n, HH, 0, stream>>>(acc, denom, bias2, h2);

    // ================= heads =================
    gemm(h2, Wnm, bnm, o_nmu, n, LL, HH, 0);
    gemm(h2, Wnv, bnv, o_nlv, n, LL, HH, 0);
    graph_head<<<1, HH, 0, stream>>>(h2, Wgm, bgm, Wgv, bgv, o_gmu, o_glv, n);
    reparam<<<(n * LL + 255) / 256, 256, 0, stream>>>(o_nmu, o_nlv, epsN, z, n * LL);

    // ================= node-feature decoder =================
    gemm(z, Wd1, bd1, bufA, n, HH, LL, 1);
    layernorm64<<<n, HH, 0, stream>>>(bufA, ln1g, ln1b);
    gemm(bufA, Wd2, bd2, bufB, n, HH, HH, 1);
    layernorm64<<<n, HH, 0, stream>>>(bufB, ln2g, ln2b);
    gemm(bufB, Wd3, bd3, nf3, n, FIN, HH, 0);

    // ================= feature refinement =================
    concat_kernel<<<(n * (FIN + HH) + 255) / 256, 256, 0, stream>>>(nf3, h2, cat, n);
    gemm(cat, Wf1, bf1, bufA, n, HH, FIN + HH, 1);
    layernorm64<<<n, HH, 0, stream>>>(bufA, lnfg, lnfb);
    gemm(bufA, Wf2, bf2, o_nf, n, FIN, HH, 0);

    // ================= all-pairs edge predictor =================
    gemm(z, We1,            be1,     bufA, n, HH, LL, 0);   // ZA = z@We1_top + be1
    gemm(z, We1 + LL * HH,  nullptr, bufB, n, HH, LL, 0);   // ZB = z@We1_bot
    edge_consts<<<1, HH, 0, stream>>>(lneg, lneb, We2, be2, csts);
    dim3 pg((n + TI - 1) / TI, (n + TJ - 1) / TJ);
    edge_pairs<<<pg, 256, 0, stream>>>(bufA, bufB, csts, o_el, n);
}